// GravNetConv_78864189489968
// MI455X (gfx1250) — compile-verified
//
#include <hip/hip_runtime.h>
#include <math.h>

typedef __attribute__((ext_vector_type(2))) float v2f;
typedef __attribute__((ext_vector_type(8))) float v8f;

#define C_IN   128
#define C_OUT  128
#define S_DIM  4
#define P_DIM  32
#define KNN    16

// ---------------------------------------------------------------------------
// Kernel A: s = x@Ws.T + bs ; h = x@Wh.T + bh ; sq = sum(s*s)
// One thread per row; weights staged in LDS.
// ---------------------------------------------------------------------------
__global__ void encode_kernel(const float* __restrict__ x,
                              const float* __restrict__ Ws,
                              const float* __restrict__ bs,
                              const float* __restrict__ Wh,
                              const float* __restrict__ bh,
                              float* __restrict__ s_out,
                              float* __restrict__ h_out,
                              float* __restrict__ sq_out,
                              int N) {
    __shared__ float lws[S_DIM * C_IN];      // 512
    __shared__ float lwh[P_DIM * C_IN];      // 4096
    __shared__ float lbs[S_DIM];
    __shared__ float lbh[P_DIM];

    int tid = threadIdx.x;
    for (int i = tid; i < S_DIM * C_IN; i += blockDim.x) lws[i] = Ws[i];
    for (int i = tid; i < P_DIM * C_IN; i += blockDim.x) lwh[i] = Wh[i];
    if (tid < S_DIM) lbs[tid] = bs[tid];
    if (tid < P_DIM) lbh[tid] = bh[tid];
    __syncthreads();

    int row = blockIdx.x * blockDim.x + tid;
    if (row >= N) return;

    float sa[S_DIM];
    float ha[P_DIM];
#pragma unroll
    for (int j = 0; j < S_DIM; ++j) sa[j] = lbs[j];
#pragma unroll
    for (int p = 0; p < P_DIM; ++p) ha[p] = lbh[p];

    const float4* xrow = (const float4*)(x + (size_t)row * C_IN);
    for (int c4 = 0; c4 < C_IN / 4; ++c4) {
        float4 xv = xrow[c4];
        float xc[4] = {xv.x, xv.y, xv.z, xv.w};
#pragma unroll
        for (int u = 0; u < 4; ++u) {
            int c = c4 * 4 + u;
            float v = xc[u];
#pragma unroll
            for (int j = 0; j < S_DIM; ++j) sa[j] = fmaf(v, lws[j * C_IN + c], sa[j]);
#pragma unroll
            for (int p = 0; p < P_DIM; ++p) ha[p] = fmaf(v, lwh[p * C_IN + c], ha[p]);
        }
    }

    float sq = 0.f;
#pragma unroll
    for (int j = 0; j < S_DIM; ++j) {
        s_out[row * S_DIM + j] = sa[j];
        sq = fmaf(sa[j], sa[j], sq);
    }
    sq_out[row] = sq;
#pragma unroll
    for (int p = 0; p < P_DIM; ++p) h_out[row * P_DIM + p] = ha[p];
}

// ---------------------------------------------------------------------------
// Kernel B: brute-force kNN in 4-D latent space.
// Each wave owns TWO 16-row query tiles (32 rows = 1 row per lane) so the
// top-k selection phase runs at full wave32 lane utilization. Per 16-j
// subtile: 2x V_WMMA_F32_16X16X4_F32 against a shared B operand.
// d2 = sq_i + sq_j - 2*gram. Top-16 kept sorted in registers per lane.
// ---------------------------------------------------------------------------
#define JCHUNK 256

__global__ __launch_bounds__(128)
void knn_kernel(const float* __restrict__ s,
                const float* __restrict__ sq,
                float* __restrict__ knn_w,
                int* __restrict__ knn_idx,
                int N) {
    __shared__ float sj[JCHUNK * S_DIM];   // staged latent rows (4 KB)
    __shared__ float sqj[JCHUNK];          // staged |s_j|^2 (1 KB)
    __shared__ float d2buf[4][32 * 17];    // per-wave transpose scratch (8.5 KB)

    const int tid  = threadIdx.x;
    const int lane = tid & 31;
    const int wave = tid >> 5;
    const int half = lane >> 4;
    const int lm   = lane & 15;

    const int i0 = blockIdx.x * 128 + wave * 32;   // 32 query rows per wave

    // A operands: 16x4 f32 matrices (two row-tiles).
    // lanes 0-15 -> K={0,1}, lanes 16-31 -> K={2,3}; lane%16 = M
    v2f a0, a1;
    {
        const float* ap0 = s + (size_t)(i0 + lm) * S_DIM + 2 * half;
        const float* ap1 = s + (size_t)(i0 + 16 + lm) * S_DIM + 2 * half;
        a0.x = ap0[0]; a0.y = ap0[1];
        a1.x = ap1[0]; a1.y = ap1[1];
    }
    // |s_i|^2 for the rows covered by this lane's C fragments (m = v + 8*half)
    float sqA0[8], sqA1[8];
#pragma unroll
    for (int v = 0; v < 8; ++v) {
        sqA0[v] = sq[i0 + v + 8 * half];
        sqA1[v] = sq[i0 + 16 + v + 8 * half];
    }

    float dist[KNN];
    int   idn[KNN];
#pragma unroll
    for (int t = 0; t < KNN; ++t) { dist[t] = 3.4e38f; idn[t] = 0; }

    float* db = &d2buf[wave][0];

    for (int jc = 0; jc < N; jc += JCHUNK) {
        __syncthreads();
        for (int t = tid; t < JCHUNK * S_DIM; t += 128) sj[t] = s[(size_t)jc * S_DIM + t];
        for (int t = tid; t < JCHUNK; t += 128) sqj[t] = sq[jc + t];
        __syncthreads();

#pragma unroll 1
        for (int sub = 0; sub < JCHUNK / 16; ++sub) {
            const int jb = sub * 16;
            // B operand: 4x16, lane carries N=lm, half-wave carries K pair
            v2f b;
            b.x = sj[(jb + lm) * S_DIM + 2 * half];
            b.y = sj[(jb + lm) * S_DIM + 2 * half + 1];
            float sqn = sqj[jb + lm];

            v8f c0 = {};
            v8f c1 = {};
            c0 = __builtin_amdgcn_wmma_f32_16x16x4_f32(
                     false, a0, false, b, (short)0, c0, false, false);
            c1 = __builtin_amdgcn_wmma_f32_16x16x4_f32(
                     false, a1, false, b, (short)0, c1, false, false);

            // d2 + transpose through LDS: element (m = v+8*half, n = lm)
#pragma unroll
            for (int v = 0; v < 8; ++v) {
                float d20 = sqA0[v] + sqn - 2.0f * c0[v];
                float d21 = sqA1[v] + sqn - 2.0f * c1[v];
                db[(v + 8 * half) * 17 + lm]      = d20;
                db[(16 + v + 8 * half) * 17 + lm] = d21;
            }

            // every lane owns one query row: full-wave top-k maintenance
            {
                const int jgl = jc + jb;
#pragma unroll
                for (int n = 0; n < 16; ++n) {
                    float d = db[lane * 17 + n];
                    if (d < dist[KNN - 1]) {
                        dist[KNN - 1] = d;
                        idn[KNN - 1]  = jgl + n;
#pragma unroll
                        for (int t = KNN - 1; t >= 1; --t) {
                            if (dist[t] < dist[t - 1]) {
                                float td = dist[t]; dist[t] = dist[t - 1]; dist[t - 1] = td;
                                int   ti = idn[t];  idn[t]  = idn[t - 1];  idn[t - 1]  = ti;
                            }
                        }
                    }
                }
            }
        }
    }

    {
        int row = i0 + lane;
#pragma unroll
        for (int t = 0; t < KNN; ++t) {
            knn_w[row * KNN + t]   = __expf(-10.0f * dist[t]);
            knn_idx[row * KNN + t] = idn[t];
        }
    }
}

// ---------------------------------------------------------------------------
// Kernel C: agg[row] = [mean_k(h[idx]*w), max_k(h[idx]*w)]   -> [N, 64]
// One wave per row; lane = P dim (coalesced h gathers).
// ---------------------------------------------------------------------------
__global__ void aggregate_kernel(const float* __restrict__ h,
                                 const float* __restrict__ knn_w,
                                 const int* __restrict__ knn_idx,
                                 float* __restrict__ agg,
                                 int N) {
    int wave = threadIdx.x >> 5;
    int lane = threadIdx.x & 31;
    int row = blockIdx.x * (blockDim.x >> 5) + wave;
    if (row >= N) return;

    float mean = 0.f;
    float mx = -3.4e38f;
#pragma unroll
    for (int t = 0; t < KNN; ++t) {
        int   id = knn_idx[row * KNN + t];
        float wv = knn_w[row * KNN + t];
        float m  = h[(size_t)id * P_DIM + lane] * wv;
        mean += m;
        mx = fmaxf(mx, m);
    }
    agg[(size_t)row * (2 * P_DIM) + lane]         = mean * (1.0f / KNN);
    agg[(size_t)row * (2 * P_DIM) + P_DIM + lane] = mx;
}

// ---------------------------------------------------------------------------
// Kernel D: out = x@W1.T + agg@W2.T + b2, tiled 16x16, f32 WMMA accumulation.
// ---------------------------------------------------------------------------
__global__ __launch_bounds__(256)
void outgemm_kernel(const float* __restrict__ x,
                    const float* __restrict__ agg,
                    const float* __restrict__ W1,
                    const float* __restrict__ W2,
                    const float* __restrict__ b2,
                    float* __restrict__ out,
                    int N) {
    const int tid  = threadIdx.x;
    const int lane = tid & 31;
    const int wave = tid >> 5;
    const int half = lane >> 4;
    const int lm   = lane & 15;

    const int tilesPerRow = C_OUT / 16;               // 8
    const int tile = blockIdx.x * 8 + wave;
    const int i0 = (tile / tilesPerRow) * 16;
    const int n0 = (tile % tilesPerRow) * 16;
    if (i0 >= N) return;

    v8f c = {};

    // K over x / W1  (K = 128)
    {
        const float* xa = x  + (size_t)(i0 + lm) * C_IN + 2 * half;
        const float* wb = W1 + (size_t)(n0 + lm) * C_IN + 2 * half;
#pragma unroll 4
        for (int k = 0; k < C_IN; k += 4) {
            v2f a; a.x = xa[k]; a.y = xa[k + 1];
            v2f b; b.x = wb[k]; b.y = wb[k + 1];
            c = __builtin_amdgcn_wmma_f32_16x16x4_f32(
                    false, a, false, b, (short)0, c, false, false);
        }
    }
    // K over agg / W2  (K = 64)
    {
        const float* ga  = agg + (size_t)(i0 + lm) * (2 * P_DIM) + 2 * half;
        const float* w2b = W2  + (size_t)(n0 + lm) * (2 * P_DIM) + 2 * half;
#pragma unroll 4
        for (int k = 0; k < 2 * P_DIM; k += 4) {
            v2f a; a.x = ga[k];  a.y = ga[k + 1];
            v2f b; b.x = w2b[k]; b.y = w2b[k + 1];
            c = __builtin_amdgcn_wmma_f32_16x16x4_f32(
                    false, a, false, b, (short)0, c, false, false);
        }
    }

    const float bias = b2[n0 + lm];
#pragma unroll
    for (int v = 0; v < 8; ++v) {
        int m = i0 + v + 8 * half;
        out[(size_t)m * C_OUT + n0 + lm] = c[v] + bias;
    }
}

// ---------------------------------------------------------------------------
extern "C" void kernel_launch(void* const* d_in, const int* in_sizes, int n_in,
                              void* d_out, int out_size, void* d_ws, size_t ws_size,
                              hipStream_t stream) {
    const float* x  = (const float*)d_in[0];
    const float* Ws = (const float*)d_in[1];
    const float* bs = (const float*)d_in[2];
    const float* Wh = (const float*)d_in[3];
    const float* bh = (const float*)d_in[4];
    const float* W1 = (const float*)d_in[5];
    const float* W2 = (const float*)d_in[6];
    const float* b2 = (const float*)d_in[7];
    float* out = (float*)d_out;

    const int N = in_sizes[0] / C_IN;   // 16384

    // workspace layout (floats)
    float* ws = (float*)d_ws;
    float* s_buf   = ws;                         // N*4
    float* sq_buf  = s_buf + (size_t)N * S_DIM;  // N
    float* h_buf   = sq_buf + N;                 // N*32
    float* w_buf   = h_buf + (size_t)N * P_DIM;  // N*16
    int*   idx_buf = (int*)(w_buf + (size_t)N * KNN);   // N*16
    float* agg_buf = (float*)(idx_buf + (size_t)N * KNN); // N*64

    // A: encode
    encode_kernel<<<(N + 255) / 256, 256, 0, stream>>>(
        x, Ws, bs, Wh, bh, s_buf, h_buf, sq_buf, N);

    // B: kNN (128 query rows per block, 4 waves x 32-row double-tiles)
    knn_kernel<<<N / 128, 128, 0, stream>>>(s_buf, sq_buf, w_buf, idx_buf, N);

    // C: aggregate (8 rows per block)
    aggregate_kernel<<<N / 8, 256, 0, stream>>>(h_buf, w_buf, idx_buf, agg_buf, N);

    // D: output GEMM (8 tiles of 16x16 per block)
    outgemm_kernel<<<(N / 16) * (C_OUT / 16) / 8, 256, 0, stream>>>(
        x, agg_buf, W1, W2, b2, out, N);
}